// RESMAv2Standard_40269613367380
// MI455X (gfx1250) — compile-verified
//
#include <hip/hip_runtime.h>
#include <hip/hip_bf16.h>

typedef float v2f __attribute__((ext_vector_type(2)));
typedef float v8f __attribute__((ext_vector_type(8)));

#define HDIM 128
#define HEADS 4
#define HD 32
#define NEG_SLOPE 0.2f
#define MT 4   // M-tiles per wave (B-fragment register reuse factor)

// ---------------- WMMA fp32 GEMM:  C[M,Nout] = A[M,K] * W[Nout,K]^T (+bias)(+relu) ----------------
// One wave32 computes MT consecutive 16x16 tiles (same N-tile) with V_WMMA_F32_16X16X4_F32,
// loading each B fragment once and reusing it across MT A fragments.
__global__ __launch_bounds__(256) void gemm_wmma_f32(
    const float* __restrict__ A, const float* __restrict__ W,
    const float* __restrict__ bias, float* __restrict__ C,
    int M, int Nout, int K, int ldc, int relu)
{
    int wavesPerBlk = blockDim.x >> 5;
    int waveId = blockIdx.x * wavesPerBlk + (threadIdx.x >> 5);
    int tilesN = Nout >> 4;
    int tilesM = M >> 4;
    int groupsM = (tilesM + MT - 1) / MT;
    if (waveId >= groupsM * tilesN) return;
    int gm = waveId / tilesN;
    int tn = waveId % tilesN;
    int tm0 = gm * MT;
    int mt = min(MT, tilesM - tm0);          // wave-uniform
    int lane = threadIdx.x & 31;
    int half = lane >> 4;
    int l16 = lane & 15;

    const float* __restrict__ brow  = W + (size_t)(tn * 16 + l16) * K + 2 * half;
    const float* __restrict__ arow0 = A + (size_t)(tm0 * 16 + l16) * K + 2 * half;
    const size_t astride = (size_t)16 * K;   // one M-tile of rows

    v8f acc[MT];
    #pragma unroll
    for (int i = 0; i < MT; ++i) acc[i] = (v8f){0.f, 0.f, 0.f, 0.f, 0.f, 0.f, 0.f, 0.f};

    if (mt == MT) {
        for (int k = 0; k < K; k += 4) {
            v2f b;
            b.x = brow[k];
            b.y = brow[k + 1];
            #pragma unroll
            for (int i = 0; i < MT; ++i) {
                const float* ar = arow0 + i * astride;
                v2f a;
                a.x = ar[k];
                a.y = ar[k + 1];
                acc[i] = __builtin_amdgcn_wmma_f32_16x16x4_f32(false, a, false, b, (short)0,
                                                               acc[i], false, false);
            }
        }
    } else {
        for (int k = 0; k < K; k += 4) {
            v2f b;
            b.x = brow[k];
            b.y = brow[k + 1];
            for (int i = 0; i < mt; ++i) {
                const float* ar = arow0 + i * astride;
                v2f a;
                a.x = ar[k];
                a.y = ar[k + 1];
                acc[i] = __builtin_amdgcn_wmma_f32_16x16x4_f32(false, a, false, b, (short)0,
                                                               acc[i], false, false);
            }
        }
    }

    int coln = tn * 16 + l16;
    float bv = bias ? bias[coln] : 0.0f;
    for (int i = 0; i < mt; ++i) {
        #pragma unroll
        for (int v = 0; v < 8; ++v) {
            int rowm = (tm0 + i) * 16 + v + 8 * half;
            float o = acc[i][v] + bv;
            if (relu) o = fmaxf(o, 0.0f);
            C[(size_t)rowm * ldc + coln] = o;
        }
    }
}

// ---------------- degree / sym-norm ----------------
__global__ void deg_kernel(const int* __restrict__ col, float* __restrict__ deg, int E) {
    int t = blockIdx.x * blockDim.x + threadIdx.x;
    if (t < E) atomicAdd(&deg[col[t]], 1.0f);
}

__global__ void dis_kernel(float* __restrict__ deg, int N) {
    int t = blockIdx.x * blockDim.x + threadIdx.x;
    if (t < N) {
        float d = deg[t];
        deg[t] = d > 0.0f ? rsqrtf(d) : 0.0f;
    }
}

// ---------------- e8: agg[row] += dis[row]*dis[col] * x[col] ----------------
__global__ void e8_agg_kernel(const int* __restrict__ row, const int* __restrict__ col,
                              const float* __restrict__ dis, const float* __restrict__ x,
                              float* __restrict__ agg, int E) {
    int t = blockIdx.x * blockDim.x + threadIdx.x;
    int e = t >> 5;
    if (e >= E) return;
    int f = (t & 31) * 4;
    int r = row[e], c = col[e];
    float v = dis[r] * dis[c];
    const float4 xv = *(const float4*)(x + (size_t)c * HDIM + f);
    float* o = agg + (size_t)r * HDIM + f;
    atomicAdd(o + 0, v * xv.x);
    atomicAdd(o + 1, v * xv.y);
    atomicAdd(o + 2, v * xv.z);
    atomicAdd(o + 3, v * xv.w);
}

// ---------------- GAT ----------------
__global__ void gat_scores_kernel(const float* __restrict__ h,
                                  const float* __restrict__ asrc, const float* __restrict__ adst,
                                  float* __restrict__ ssrc, float* __restrict__ sdst, int N) {
    int t = blockIdx.x * blockDim.x + threadIdx.x;
    if (t >= N * HEADS) return;
    int n = t >> 2, hd = t & 3;
    const float* hv = h + (size_t)n * HDIM + hd * HD;
    const float* as = asrc + hd * HD;
    const float* ad = adst + hd * HD;
    float s = 0.f, d = 0.f;
    #pragma unroll
    for (int j = 0; j < HD; ++j) {
        float v = hv[j];
        s += v * as[j];
        d += v * ad[j];
    }
    ssrc[t] = s;
    sdst[t] = d;
}

__device__ __forceinline__ unsigned f2key(float f) {
    unsigned u = __float_as_uint(f);
    return (u & 0x80000000u) ? ~u : (u | 0x80000000u);
}
__device__ __forceinline__ float key2f(unsigned k) {
    unsigned u = (k & 0x80000000u) ? (k & 0x7FFFFFFFu) : ~k;
    return __uint_as_float(u);
}
__device__ __forceinline__ float lrelu(float x) { return x > 0.f ? x : NEG_SLOPE * x; }

__global__ void gat_max_kernel(const int* __restrict__ row, const int* __restrict__ col,
                               const float* __restrict__ ssrc, const float* __restrict__ sdst,
                               unsigned* __restrict__ mk, int E, int N) {
    int t = blockIdx.x * blockDim.x + threadIdx.x;
    if (t >= (E + N) * HEADS) return;
    int e = t >> 2, hd = t & 3;
    int s, d;
    if (e < E) { s = row[e]; d = col[e]; } else { s = e - E; d = s; }
    float sc = lrelu(ssrc[s * HEADS + hd] + sdst[d * HEADS + hd]);
    atomicMax(&mk[d * HEADS + hd], f2key(sc));
}

__global__ void gat_den_kernel(const int* __restrict__ row, const int* __restrict__ col,
                               const float* __restrict__ ssrc, const float* __restrict__ sdst,
                               const unsigned* __restrict__ mk, float* __restrict__ den,
                               int E, int N) {
    int t = blockIdx.x * blockDim.x + threadIdx.x;
    if (t >= (E + N) * HEADS) return;
    int e = t >> 2, hd = t & 3;
    int s, d;
    if (e < E) { s = row[e]; d = col[e]; } else { s = e - E; d = s; }
    float sc = lrelu(ssrc[s * HEADS + hd] + sdst[d * HEADS + hd]);
    float m = key2f(mk[d * HEADS + hd]);
    atomicAdd(&den[d * HEADS + hd], expf(sc - m));
}

// out (right half of cat buffer, row stride 256): out[dst] += alpha * h[src]
__global__ void gat_agg_kernel(const int* __restrict__ row, const int* __restrict__ col,
                               const float* __restrict__ ssrc, const float* __restrict__ sdst,
                               const unsigned* __restrict__ mk, const float* __restrict__ den,
                               const float* __restrict__ h, float* __restrict__ cat,
                               int E, int N) {
    int t = blockIdx.x * blockDim.x + threadIdx.x;
    int e = t >> 5;
    if (e >= E + N) return;
    int f = (t & 31) * 4;
    int hd = f >> 5;
    int s, d;
    if (e < E) { s = row[e]; d = col[e]; } else { s = e - E; d = s; }
    float sc = lrelu(ssrc[s * HEADS + hd] + sdst[d * HEADS + hd]);
    float m = key2f(mk[d * HEADS + hd]);
    float alpha = expf(sc - m) / (den[d * HEADS + hd] + 1e-16f);
    const float4 hv = *(const float4*)(h + (size_t)s * HDIM + f);
    float* o = cat + (size_t)d * (2 * HDIM) + HDIM + f;
    atomicAdd(o + 0, alpha * hv.x);
    atomicAdd(o + 1, alpha * hv.y);
    atomicAdd(o + 2, alpha * hv.z);
    atomicAdd(o + 3, alpha * hv.w);
}

__global__ void add_gb_kernel(float* __restrict__ cat, const float* __restrict__ gb, int N) {
    int t = blockIdx.x * blockDim.x + threadIdx.x;
    if (t >= N * HDIM) return;
    int n = t >> 7, j = t & 127;
    cat[(size_t)n * (2 * HDIM) + HDIM + j] += gb[j];
}

// ---------------- x = relu(layernorm(y + res)) ; res/out in-place in x ----------------
__global__ __launch_bounds__(256) void ln_relu_kernel(const float* __restrict__ y,
                                                      float* __restrict__ x,
                                                      const float* __restrict__ g,
                                                      const float* __restrict__ be, int N) {
    int wid = blockIdx.x * (blockDim.x >> 5) + (threadIdx.x >> 5);
    if (wid >= N) return;
    int lane = threadIdx.x & 31;
    size_t base = (size_t)wid * HDIM + lane * 4;
    float4 v = *(const float4*)(y + base);
    const float4 r = *(const float4*)(x + base);
    v.x += r.x; v.y += r.y; v.z += r.z; v.w += r.w;
    float s = v.x + v.y + v.z + v.w;
    float s2 = v.x * v.x + v.y * v.y + v.z * v.z + v.w * v.w;
    #pragma unroll
    for (int m = 16; m >= 1; m >>= 1) {
        s += __shfl_xor(s, m, 32);
        s2 += __shfl_xor(s2, m, 32);
    }
    float mean = s * (1.0f / HDIM);
    float var = s2 * (1.0f / HDIM) - mean * mean;
    float inv = rsqrtf(var + 1e-5f);
    const float4 gg = *(const float4*)(g + lane * 4);
    const float4 bb = *(const float4*)(be + lane * 4);
    float4 o;
    o.x = fmaxf((v.x - mean) * inv * gg.x + bb.x, 0.f);
    o.y = fmaxf((v.y - mean) * inv * gg.y + bb.y, 0.f);
    o.z = fmaxf((v.z - mean) * inv * gg.z + bb.z, 0.f);
    o.w = fmaxf((v.w - mean) * inv * gg.w + bb.w, 0.f);
    *(float4*)(x + base) = o;
}

// ---------------- out[n] = sigmoid(r1[n,:] . rw2 + rb2) ----------------
__global__ __launch_bounds__(256) void readout2_kernel(const float* __restrict__ r1,
                                                       const float* __restrict__ w,
                                                       const float* __restrict__ b,
                                                       float* __restrict__ out, int N) {
    int wid = blockIdx.x * (blockDim.x >> 5) + (threadIdx.x >> 5);
    if (wid >= N) return;
    int lane = threadIdx.x & 31;
    const float* rr = r1 + (size_t)wid * 64;
    float s = rr[lane] * w[lane] + rr[32 + lane] * w[32 + lane];
    #pragma unroll
    for (int m = 16; m >= 1; m >>= 1) s += __shfl_xor(s, m, 32);
    if (lane == 0) out[wid] = 1.0f / (1.0f + expf(-(s + b[0])));
}

// ======================================================================================
extern "C" void kernel_launch(void* const* d_in, const int* in_sizes, int n_in,
                              void* d_out, int out_size, void* d_ws, size_t ws_size,
                              hipStream_t stream) {
    const int N = in_sizes[0] / HDIM;
    const int E = in_sizes[1] / 2;

    const float* x_in  = (const float*)d_in[0];
    const int*   ei    = (const int*)d_in[1];
    const int*   row   = ei;
    const int*   col   = ei + E;
    const float* W_emb = (const float*)d_in[2];
    const float* b_emb = (const float*)d_in[3];
    const float* e8w[2]  = {(const float*)d_in[4],  (const float*)d_in[13]};
    const float* gw[2]   = {(const float*)d_in[5],  (const float*)d_in[14]};
    const float* asrc[2] = {(const float*)d_in[6],  (const float*)d_in[15]};
    const float* adst[2] = {(const float*)d_in[7],  (const float*)d_in[16]};
    const float* gb[2]   = {(const float*)d_in[8],  (const float*)d_in[17]};
    const float* fw[2]   = {(const float*)d_in[9],  (const float*)d_in[18]};
    const float* fb[2]   = {(const float*)d_in[10], (const float*)d_in[19]};
    const float* gam[2]  = {(const float*)d_in[11], (const float*)d_in[20]};
    const float* bet[2]  = {(const float*)d_in[12], (const float*)d_in[21]};
    const float* rw1 = (const float*)d_in[22];
    const float* rb1 = (const float*)d_in[23];
    const float* rw2 = (const float*)d_in[24];
    const float* rb2 = (const float*)d_in[25];
    float* out = (float*)d_out;

    // workspace carve-up
    size_t off = 0;
    auto alloc = [&](size_t bytes) -> void* {
        void* p = (char*)d_ws + off;
        off += bytes;
        off = (off + 255) & ~(size_t)255;
        return p;
    };
    float*    B_x   = (float*)alloc((size_t)N * HDIM * 4);      // node features
    float*    B_cat = (float*)alloc((size_t)N * 2 * HDIM * 4);  // [xe | xg]
    float*    B_tmp = (float*)alloc((size_t)N * HDIM * 4);      // agg -> h -> y
    float*    B_r1  = (float*)alloc((size_t)N * 64 * 4);
    float*    B_dis = (float*)alloc((size_t)N * 4);
    float*    B_ss  = (float*)alloc((size_t)N * HEADS * 4);
    float*    B_sd  = (float*)alloc((size_t)N * HEADS * 4);
    unsigned* B_mk  = (unsigned*)alloc((size_t)N * HEADS * 4);
    float*    B_den = (float*)alloc((size_t)N * HEADS * 4);

    auto cdiv = [](long a, long b) { return (int)((a + b - 1) / b); };
    auto gemm = [&](const float* A, const float* W, const float* bias, float* C,
                    int M, int Nout, int K, int ldc, int relu) {
        int tilesM = M / 16;
        int tiles = ((tilesM + MT - 1) / MT) * (Nout / 16);
        gemm_wmma_f32<<<cdiv(tiles, 8), 256, 0, stream>>>(A, W, bias, C, M, Nout, K, ldc, relu);
    };

    // degree / sym-norm (edge structure identical for both layers)
    hipMemsetAsync(B_dis, 0, (size_t)N * 4, stream);
    deg_kernel<<<cdiv(E, 256), 256, 0, stream>>>(col, B_dis, E);
    dis_kernel<<<cdiv(N, 256), 256, 0, stream>>>(B_dis, N);

    // embedding: x = x_in @ W_emb^T + b_emb
    gemm(x_in, W_emb, b_emb, B_x, N, HDIM, HDIM, HDIM, 0);

    for (int L = 0; L < 2; ++L) {
        hipMemsetAsync(B_tmp, 0, (size_t)N * HDIM * 4, stream);
        hipMemsetAsync(B_cat, 0, (size_t)N * 2 * HDIM * 4, stream);
        hipMemsetAsync(B_mk, 0, (size_t)N * HEADS * 4, stream);
        hipMemsetAsync(B_den, 0, (size_t)N * HEADS * 4, stream);

        // e8 branch: agg then GEMM into left half of cat
        e8_agg_kernel<<<cdiv((long)E * 32, 256), 256, 0, stream>>>(row, col, B_dis, B_x, B_tmp, E);
        gemm(B_tmp, e8w[L], nullptr, B_cat, N, HDIM, HDIM, 2 * HDIM, 0);

        // GAT branch: h = x @ gw^T (reuse B_tmp), scores, 3-pass edge softmax, scatter
        gemm(B_x, gw[L], nullptr, B_tmp, N, HDIM, HDIM, HDIM, 0);
        gat_scores_kernel<<<cdiv((long)N * HEADS, 256), 256, 0, stream>>>(B_tmp, asrc[L], adst[L],
                                                                          B_ss, B_sd, N);
        gat_max_kernel<<<cdiv((long)(E + N) * HEADS, 256), 256, 0, stream>>>(row, col, B_ss, B_sd,
                                                                             B_mk, E, N);
        gat_den_kernel<<<cdiv((long)(E + N) * HEADS, 256), 256, 0, stream>>>(row, col, B_ss, B_sd,
                                                                             B_mk, B_den, E, N);
        gat_agg_kernel<<<cdiv((long)(E + N) * 32, 256), 256, 0, stream>>>(row, col, B_ss, B_sd,
                                                                          B_mk, B_den, B_tmp,
                                                                          B_cat, E, N);
        add_gb_kernel<<<cdiv((long)N * HDIM, 256), 256, 0, stream>>>(B_cat, gb[L], N);

        // fusion: y = cat @ fw^T + fb   (reuse B_tmp as y; h is dead now)
        gemm(B_cat, fw[L], fb[L], B_tmp, N, HDIM, 2 * HDIM, HDIM, 0);

        // x = relu(LN(y + x))
        ln_relu_kernel<<<cdiv((long)N * 32, 256), 256, 0, stream>>>(B_tmp, B_x, gam[L], bet[L], N);
    }

    // readout
    gemm(B_x, rw1, rb1, B_r1, N, 64, HDIM, 64, 1);
    readout2_kernel<<<cdiv((long)N * 32, 256), 256, 0, stream>>>(B_r1, rw2, rb2, out, N);
}